// MVPMLoss_30356828848315
// MI455X (gfx1250) — compile-verified
//
#include <hip/hip_runtime.h>
#include <hip/hip_bf16.h>

// ---------------------------------------------------------------------------
// MVPM multi-term loss, MI455X (gfx1250, wave32).
// Memory-bound: ~265 MB streamed -> ~11.4 us floor at 23.3 TB/s.
// Stage 1: three streaming pass kernels (coalesced b32 loads, register accums,
//          wave32 shuffle reduce, block partials to workspace).
// Stage 2: cross-block reduction via V_WMMA_F32_16X16X4_F32 (ones x partials).
// Stage 3: tiny finalize kernel producing the 7 loss scalars.
// Deterministic (no float atomics) -> graph replays are bit-stable.
// ---------------------------------------------------------------------------

#define HW   65536     // 256*256
#define NCHK 64        // chunks per batch (1024 px each)
#define NBLK 256       // 4 batches * 64 chunks
#define SLOTS 128      // partial slots per block

typedef __attribute__((ext_vector_type(2))) float v2f;
typedef __attribute__((ext_vector_type(8))) float v8f;

__device__ __forceinline__ float fsigmoid(float x) {
    return 1.0f / (1.0f + __expf(-x));
}
__device__ __forceinline__ float flogsig(float x) {
    // log(sigmoid(x)) = min(x,0) - log(1+exp(-|x|)), stable
    return fminf(x, 0.0f) - __logf(1.0f + __expf(-fabsf(x)));
}
__device__ __forceinline__ float wave_sum(float v) {
    v += __shfl_down(v, 16, 32);
    v += __shfl_down(v, 8, 32);
    v += __shfl_down(v, 4, 32);
    v += __shfl_down(v, 2, 32);
    v += __shfl_down(v, 1, 32);
    return v;
}

// ---------------------------------------------------------------------------
// Slot map (per block row of `part`):
//   0: bce sum      1: nocs masked-dn   2: nocs dn    3: nocs cnt
//   4: loc mdn      5: loc dn           6: loc cnt
//   7: rot mdn      8: rot dn           9: rot cnt
//  10: skin CE sum  11..15: zero pad
//  16..31 : denom[j]  = sum m*sig(js_j)
//  32..79 : loc numer = sum sig(loc_c)*m^2*sig(js_{c/3})
//  80..127: rot numer
// ---------------------------------------------------------------------------

// Kernel 1: BCE + NOCS masked L2 + skin cross-entropy  (27 floats/pixel)
__global__ __launch_bounds__(256)
void k_pass_light(const float* __restrict__ outp, const float* __restrict__ tarm,
                  float* __restrict__ part) {
    const int blk = blockIdx.x;
    const int b = blk >> 6, chunk = blk & 63;
    const int tid = threadIdx.x;
    const float* ob = outp + (size_t)b * 134 * HW;
    const float* tb = tarm + (size_t)b * 101 * HW;

    float a0 = 0.f, a1 = 0.f, a2 = 0.f, a3 = 0.f, a4 = 0.f;
    #pragma unroll
    for (int k = 0; k < 4; ++k) {
        const int p = chunk * 1024 + k * 256 + tid;
        const float m = tb[3 * HW + p];
        const float x = ob[3 * HW + p];
        a0 += m * flogsig(x) + (1.0f - m) * flogsig(-x);

        float d2 = 0.f;
        #pragma unroll
        for (int c = 0; c < 3; ++c) {
            const float d = ob[c * HW + p] - tb[c * HW + p];
            d2 += d * d;
        }
        const float dn = sqrtf(d2);
        a2 += dn;
        if (m > 0.7f && dn != 0.0f) { a1 += dn; a3 += 1.0f; }

        int label = (int)tb[100 * HW + p];
        label = label < 0 ? 0 : (label > 17 ? 17 : label);
        float z[18];
        float zmax = -3.4e38f, zlab = 0.0f;
        #pragma unroll
        for (int kk = 0; kk < 18; ++kk) {
            z[kk] = ob[(100 + kk) * HW + p];
            zmax = fmaxf(zmax, z[kk]);
            if (kk == label) zlab = z[kk];
        }
        float se = 0.f;
        #pragma unroll
        for (int kk = 0; kk < 18; ++kk) se += __expf(z[kk] - zmax);
        a4 += __logf(se) + zmax - zlab;
    }

    __shared__ float lds[8 * 8];
    const int wave = tid >> 5, lane = tid & 31;
    float acc[5] = {a0, a1, a2, a3, a4};
    #pragma unroll
    for (int i = 0; i < 5; ++i) {
        const float v = wave_sum(acc[i]);
        if (lane == 0) lds[wave * 8 + i] = v;
    }
    __syncthreads();
    if (tid < 5) {
        float s = 0.f;
        #pragma unroll
        for (int w = 0; w < 8; ++w) s += lds[w * 8 + tid];
        const int slot = (tid < 4) ? tid : 10;
        part[blk * SLOTS + slot] = s;
    } else if (tid < 10) {
        part[blk * SLOTS + 6 + tid] = 0.0f;   // slots 11..15 zero
    }
}

// Kernel 2/3: 48-channel map masked-L2 + pose-weighted sums (113 floats/pixel)
template <int CH0, bool WITH_DENOM>
__global__ __launch_bounds__(256)
void k_pass_maps(const float* __restrict__ outp, const float* __restrict__ tarm,
                 float* __restrict__ part) {
    constexpr int NACC = WITH_DENOM ? 67 : 51;    // 3 + (16) + 48
    constexpr int NUMB = WITH_DENOM ? 19 : 3;
    const int blk = blockIdx.x;
    const int b = blk >> 6, chunk = blk & 63;
    const int tid = threadIdx.x;
    const float* ob = outp + (size_t)b * 134 * HW;
    const float* tb = tarm + (size_t)b * 101 * HW;

    float acc[NACC];
    #pragma unroll
    for (int i = 0; i < NACC; ++i) acc[i] = 0.0f;

    #pragma unroll
    for (int k = 0; k < 4; ++k) {
        const int p = chunk * 1024 + k * 256 + tid;
        const float m = tb[3 * HW + p];
        const float m2 = m * m;
        float w[16];
        #pragma unroll
        for (int j = 0; j < 16; ++j) {
            const float s = fsigmoid(ob[(118 + j) * HW + p]);
            if (WITH_DENOM) acc[3 + j] += m * s;    // denom: sum m*sig(js)
            w[j] = m2 * s;                           // numer weight
        }
        float d2 = 0.f;
        #pragma unroll
        for (int c = 0; c < 48; ++c) {
            const float pc = fsigmoid(ob[(CH0 + c) * HW + p]);
            const float tc = fsigmoid(tb[(CH0 + c) * HW + p]);
            const float d = pc - tc;
            d2 += d * d;
            acc[NUMB + c] += pc * w[c / 3];
        }
        const float dn = sqrtf(d2);
        acc[1] += dn;
        if (m > 0.7f && dn != 0.0f) { acc[0] += dn; acc[2] += 1.0f; }
    }

    __shared__ float lds[8 * 68];
    const int wave = tid >> 5, lane = tid & 31;
    #pragma unroll
    for (int i = 0; i < NACC; ++i) {
        const float v = wave_sum(acc[i]);
        if (lane == 0) lds[wave * 68 + i] = v;
    }
    __syncthreads();
    if (tid < NACC) {
        float s = 0.f;
        #pragma unroll
        for (int w2 = 0; w2 < 8; ++w2) s += lds[w2 * 68 + tid];
        int slot;
        if (WITH_DENOM)
            slot = (tid < 3) ? 4 + tid : ((tid < 19) ? 16 + (tid - 3) : 32 + (tid - 19));
        else
            slot = (tid < 3) ? 7 + tid : 80 + (tid - 3);
        part[blk * SLOTS + slot] = s;
    }
}

// Stage 2: per-batch reduction of 64 block rows x 128 slots using
// V_WMMA_F32_16X16X4_F32 with A = ones(16x4): D[m][n] = sum_k B[k][n].
// The ones-matrix makes the K assignment symmetric, so any bijective packing
// of the 4 block-rows into the (vgpr, lane-half) slots of each column is valid.
__global__ __launch_bounds__(256)
void k_reduce_wmma(const float* __restrict__ part, float* __restrict__ S) {
    const int tid = threadIdx.x;
    const int wave = tid >> 5, lane = tid & 31;
    const int n = lane & 15, half = lane >> 4;
    const int slot = wave * 16 + n;           // 8 waves * 16 slots = 128

    v2f a;
    a.x = 1.0f; a.y = 1.0f;                   // ones 16x4 A-matrix

    #pragma unroll
    for (int b = 0; b < 4; ++b) {
        v8f c = {0.f, 0.f, 0.f, 0.f, 0.f, 0.f, 0.f, 0.f};
        for (int kc = 0; kc < 64; kc += 4) {
            const int row = b * 64 + kc + half * 2;
            v2f bm;
            bm.x = part[(size_t)row * SLOTS + slot];
            bm.y = part[(size_t)(row + 1) * SLOTS + slot];
            c = __builtin_amdgcn_wmma_f32_16x16x4_f32(
                    /*neg_a=*/false, a, /*neg_b=*/false, bm,
                    /*c_mod=*/(short)0, c, /*reuse_a=*/false, /*reuse_b=*/false);
        }
        // all rows of D are identical column sums; take row M=0 (lanes 0..15)
        if (half == 0) S[b * SLOTS + slot] = c[0];
    }
}

// Stage 3: compute the 7 scalar losses.
__global__ void k_finalize(const float* __restrict__ S,
                           const float* __restrict__ tpose,
                           float* __restrict__ res) {
    if (threadIdx.x != 0 || blockIdx.x != 0) return;
    const float N = (float)HW;
    float s_bce = 0.f, s_ce = 0.f;
    float nocs = 0.f, locm = 0.f, rotm = 0.f, locl = 0.f, rotl = 0.f;
    for (int b = 0; b < 4; ++b) {
        const float* Sb = S + b * SLOTS;
        s_bce += Sb[0];
        s_ce  += Sb[10];
        nocs += (Sb[3] > 0.5f) ? Sb[1] / Sb[3] : Sb[2] * (1.0f / N);
        locm += (Sb[6] > 0.5f) ? Sb[4] / Sb[6] : Sb[5] * (1.0f / N);
        rotm += (Sb[9] > 0.5f) ? Sb[7] / Sb[9] : Sb[8] * (1.0f / N);
        for (int j = 0; j < 16; ++j) {
            const float inv = 1.0f / (Sb[16 + j] + 1e-5f);
            float dl = 0.f, dr = 0.f;
            for (int c = 0; c < 3; ++c) {
                const float pl = Sb[32 + j * 3 + c] * inv;
                const float tl = fsigmoid(tpose[(b * 16 + j) * 6 + c]);
                dl += (pl - tl) * (pl - tl);
                const float pr = Sb[80 + j * 3 + c] * inv;
                const float tr = fsigmoid(tpose[(b * 16 + j) * 6 + 3 + c]);
                dr += (pr - tr) * (pr - tr);
            }
            locl += sqrtf(dl);
            rotl += sqrtf(dr);
        }
    }
    res[0] = nocs * 0.25f;            // nocs_loss
    res[1] = -s_bce / (4.0f * N);     // mask_loss
    res[2] = locl * (1.0f / 64.0f);   // loc_loss
    res[3] = locm * 0.25f;            // loc_map_loss
    res[4] = rotl * (1.0f / 64.0f);   // rot_loss
    res[5] = rotm * 0.25f;            // rot_map_loss
    res[6] = s_ce / (4.0f * N);       // skin_loss
}

extern "C" void kernel_launch(void* const* d_in, const int* in_sizes, int n_in,
                              void* d_out, int out_size, void* d_ws, size_t ws_size,
                              hipStream_t stream) {
    (void)in_sizes; (void)n_in; (void)out_size; (void)ws_size;
    const float* outp  = (const float*)d_in[0];   // (4,134,256,256) f32
    const float* tarm  = (const float*)d_in[1];   // (4,101,256,256) f32
    const float* tpose = (const float*)d_in[2];   // (4,16,6) f32
    float* part = (float*)d_ws;                   // NBLK * SLOTS floats (128 KB)
    float* S    = part + (size_t)NBLK * SLOTS;    // 4 * SLOTS floats
    float* res  = (float*)d_out;                  // 7 floats

    hipLaunchKernelGGL(k_pass_light, dim3(NBLK), dim3(256), 0, stream, outp, tarm, part);
    hipLaunchKernelGGL((k_pass_maps<4, true>),  dim3(NBLK), dim3(256), 0, stream, outp, tarm, part);
    hipLaunchKernelGGL((k_pass_maps<52, false>), dim3(NBLK), dim3(256), 0, stream, outp, tarm, part);
    hipLaunchKernelGGL(k_reduce_wmma, dim3(1), dim3(256), 0, stream, part, S);
    hipLaunchKernelGGL(k_finalize, dim3(1), dim3(32), 0, stream, S, tpose, res);
}